// MM_Net_74680891343364
// MI455X (gfx1250) — compile-verified
//
#include <hip/hip_runtime.h>
#include <math.h>

// ---------------- problem constants ----------------
#define BSZ   4
#define CINC  64
#define COUTC 64
#define HH    128
#define WWI   128
#define KKC   9      // K
#define DMC   9
#define DSC   16
#define DCC   4
#define DIC   18
#define LLC   (HH*WWI)   // 16384

typedef float v2f __attribute__((ext_vector_type(2)));
typedef float v8f __attribute__((ext_vector_type(8)));

__device__ __forceinline__ float softplus_f(float t) {
  return (t > 20.f) ? t : log1pf(__expf(t));
}
__device__ __forceinline__ float silu_f(float t) {
  return t / (1.f + __expf(-t));
}
__device__ __forceinline__ v8f wmma4(v2f a, v2f b, v8f c) {
  return __builtin_amdgcn_wmma_f32_16x16x4_f32(false, a, false, b,
                                               (short)0, c, false, false);
}
__device__ __forceinline__ int clampi(int v, int lo, int hi) {
  return v < lo ? lo : (v > hi ? hi : v);
}

// =====================================================================
// K0a: weight re-layout.
//  wt1[oc(32, zero-padded)][tap(9)][ic(64)]  from offset_w (18,64,3,3)
//  wt2[oc(64)][tap(9)][ic(64)]               from dsc_w    (64,64,9,1)
// =====================================================================
__global__ void prep_w_k(const float* __restrict__ w1, const float* __restrict__ w2,
                         float* __restrict__ wt1, float* __restrict__ wt2) {
  const int i = blockIdx.x * blockDim.x + threadIdx.x;
  if (i < 32 * 9 * 64) {
    const int ic = i & 63, tap = (i >> 6) % 9, oc = i / (9 * 64);
    wt1[i] = (oc < 2 * KKC) ? w1[(oc * CINC + ic) * KKC + tap] : 0.f;
  }
  if (i < 64 * 9 * 64) {
    const int ic = i & 63, tap = (i >> 6) % 9, oc = i / (9 * 64);
    wt2[i] = w2[(oc * CINC + ic) * KKC + tap];
  }
}

// =====================================================================
// K0b: transpose x (B,C,H,W) -> xt (B, H*W, C), LDS-tiled, both sides
// coalesced. Makes every A-operand K-pair a single contiguous b64 load.
// =====================================================================
__global__ void transpose_x_k(const float* __restrict__ x, float* __restrict__ xt) {
  __shared__ float tile[64][65];
  const int b  = blockIdx.x >> 8;            // 256 tiles of 64 pixels / batch
  const int p0 = (blockIdx.x & 255) * 64;
  const int tx = threadIdx.x & 63;
  const int ty = threadIdx.x >> 6;           // 0..3
  for (int cc = ty; cc < 64; cc += 4)
    tile[cc][tx] = x[(((size_t)(b * CINC + cc)) << 14) + p0 + tx];
  __syncthreads();
  for (int cc = ty; cc < 64; cc += 4)
    xt[(((size_t)(b << 14)) + p0 + cc) * 64 + tx] = tile[tx][cc];
}

// =====================================================================
// K1: 3x3 conv (64 -> 18ch), pad 1, implicit GEMM on V_WMMA_F32_16X16X4.
// Per wave: 2 M-tiles (32 pixels) x 16 oc. A from pixel-major xt: one
// masked b64 per tile per step; B one b64.
// =====================================================================
__global__ void conv3x3_wmma(const float* __restrict__ xt,   // [B][16384][64]
                             const float* __restrict__ wt,   // [32][9][64]
                             const float* __restrict__ bias,
                             float* __restrict__ out) {
  const int lane = threadIdx.x & 31;
  const int ocb  = (threadIdx.x >> 5) * 16;
  const int p0   = blockIdx.x * 32;
  const int b    = p0 >> 14;
  const int rem0 = p0 & 16383;
  const int m    = lane & 15;
  const int kb   = (lane >> 4) * 2;
  const int oc   = ocb + (lane & 15);

  const int pix0 = rem0 + m,       pix1 = rem0 + 16 + m;
  const int h0 = pix0 >> 7, w0 = pix0 & 127;
  const int h1 = pix1 >> 7, w1 = pix1 & 127;

  const float* xb = xt + (((size_t)b) << 14) * 64;

  v8f acc0 = {}, acc1 = {};
  for (int tap = 0; tap < 9; ++tap) {
    const int kh = tap / 3 - 1, kw = tap % 3 - 1;
    const int hh0 = h0 + kh, ww0 = w0 + kw;
    const int hh1 = h1 + kh, ww1 = w1 + kw;
    const float m0 = (hh0 >= 0 && hh0 < HH && ww0 >= 0 && ww0 < WWI) ? 1.f : 0.f;
    const float m1 = (hh1 >= 0 && hh1 < HH && ww1 >= 0 && ww1 < WWI) ? 1.f : 0.f;
    const int o0 = clampi(hh0, 0, HH - 1) * WWI + clampi(ww0, 0, WWI - 1);
    const int o1 = clampi(hh1, 0, HH - 1) * WWI + clampi(ww1, 0, WWI - 1);
    const float* xp0 = xb + (((size_t)o0) << 6) + kb;
    const float* xp1 = xb + (((size_t)o1) << 6) + kb;
    const float* wp  = wt + ((oc * 9 + tap) << 6) + kb;
    #pragma unroll 4
    for (int is = 0; is < 16; ++is) {
      v2f a0 = *(const v2f*)xp0;
      v2f a1 = *(const v2f*)xp1;
      v2f bv = *(const v2f*)wp;
      a0.x *= m0; a0.y *= m0;
      a1.x *= m1; a1.y *= m1;
      acc0 = wmma4(a0, bv, acc0);
      acc1 = wmma4(a1, bv, acc1);
      xp0 += 4; xp1 += 4; wp += 4;
    }
  }
  // D layout: vgpr j -> M = j + (lane>=16)*8 ; N = lane&15
  const int mb  = (lane >> 4) * 8;
  const int ocw = ocb + (lane & 15);
  if (ocw < 2 * KKC) {
    const float bb = bias[ocw];
    float* ob = out + (size_t)(b * (2 * KKC) + ocw) * LLC + rem0 + mb;
    for (int j = 0; j < 8; ++j) {
      ob[j]      = acc0[j] + bb;
      ob[16 + j] = acc1[j] + bb;
    }
  }
}

// =====================================================================
// GroupNorm (optionally + tanh). One block per (batch, group).
// =====================================================================
__global__ void groupnorm_k(const float* __restrict__ in, float* __restrict__ out,
                            const float* __restrict__ gamma,
                            const float* __restrict__ beta,
                            int groups, int chPerG, int applyTanh) {
  __shared__ float s_sum[256];
  __shared__ float s_sq[256];
  const int b = blockIdx.x / groups;
  const int g = blockIdx.x % groups;
  const int N = chPerG * LLC;
  const float* base = in + ((size_t)b * groups * chPerG + (size_t)g * chPerG) * LLC;

  float su = 0.f, sq = 0.f;
  for (int i = threadIdx.x; i < N; i += blockDim.x) {
    float v = base[i]; su += v; sq += v * v;
  }
  s_sum[threadIdx.x] = su; s_sq[threadIdx.x] = sq;
  __syncthreads();
  for (int off = 128; off > 0; off >>= 1) {
    if ((int)threadIdx.x < off) {
      s_sum[threadIdx.x] += s_sum[threadIdx.x + off];
      s_sq[threadIdx.x]  += s_sq[threadIdx.x + off];
    }
    __syncthreads();
  }
  const float mean = s_sum[0] / (float)N;
  const float var  = s_sq[0] / (float)N - mean * mean;
  const float inv  = rsqrtf(var + 1e-5f);
  float* ob = out + ((size_t)b * groups * chPerG + (size_t)g * chPerG) * LLC;
  for (int i = threadIdx.x; i < N; i += blockDim.x) {
    const int c = g * chPerG + i / LLC;
    float v = (base[i] - mean) * inv * gamma[c] + beta[c];
    ob[i] = applyTanh ? tanhf(v) : v;
  }
}

// =====================================================================
// K3: channel-cumulative offsets + build Mamba sequence (trc_flatten).
// =====================================================================
__global__ void cum_seq_k(const float* __restrict__ offt,
                          float* __restrict__ ycum,
                          float* __restrict__ yseq) {
  const int i = blockIdx.x * blockDim.x + threadIdx.x;   // (b,h,w)
  const int v = i & 127;
  const int h = (i >> 7) & 127;
  const int b = i >> 14;
  float yo[KKC], rows[KKC];
  for (int c = 0; c < KKC; ++c)
    yo[c] = offt[((size_t)(b * 2 * KKC + c) * LLC) + (h << 7) + v];
  rows[4] = 0.f;
  for (int q = 1; q <= 4; ++q) {
    rows[4 + q] = rows[4 + q - 1] + yo[4 + q];
    rows[4 - q] = rows[4 - q + 1] + yo[4 - q];
  }
  const int h2 = h >> 1, t = h & 1;
  const int l = ((h2 * WWI + v) << 1) + t;
  for (int c = 0; c < KKC; ++c) {
    ycum[((size_t)(b * KKC + c) * LLC) + (h << 7) + v] = rows[c];
    yseq[((size_t)b * LLC + l) * DMC + c] = yo[c];
  }
}

// K4: xz = yseq @ in_proj_w.T   (B*L,9)x(9,36)
__global__ void in_proj_k(const float* __restrict__ ys, const float* __restrict__ w,
                          float* __restrict__ xz, int total) {
  const int i = blockIdx.x * blockDim.x + threadIdx.x;
  if (i >= total) return;
  const int j = i % 36; const int bl = i / 36;
  const float* yp = ys + (size_t)bl * DMC;
  const float* wp = w + j * DMC;
  float acc = 0.f;
  for (int c = 0; c < DMC; ++c) acc += yp[c] * wp[c];
  xz[i] = acc;
}

// K5: depthwise causal conv1d (DC=4) + SiLU
__global__ void conv1d_silu_k(const float* __restrict__ xz,
                              const float* __restrict__ cw,
                              const float* __restrict__ cb,
                              float* __restrict__ xi2, int total) {
  const int i = blockIdx.x * blockDim.x + threadIdx.x;
  if (i >= total) return;
  const int d = i % DIC; const int bl = i / DIC;
  const int b = bl / LLC; const int l = bl % LLC;
  float acc = cb[d];
  for (int j = 0; j < DCC; ++j) {
    const int ls = l - (DCC - 1) + j;
    if (ls >= 0) acc += xz[((size_t)b * LLC + ls) * 36 + d] * cw[d * DCC + j];
  }
  xi2[i] = silu_f(acc);
}

// K6: dbl = xi2 @ x_proj_w.T  (B*L,18)x(18,33)
__global__ void x_proj_k(const float* __restrict__ xi2, const float* __restrict__ w,
                         float* __restrict__ dbl, int total) {
  const int i = blockIdx.x * blockDim.x + threadIdx.x;
  if (i >= total) return;
  const int j = i % 33; const int bl = i / 33;
  const float* xp = xi2 + (size_t)bl * DIC;
  const float* wp = w + j * DIC;
  float acc = 0.f;
  for (int d = 0; d < DIC; ++d) acc += xp[d] * wp[d];
  dbl[i] = acc;
}

// K6b: dt = softplus(dbl[:,0:1] @ dt_proj_w.T + dt_proj_b)
__global__ void dt_k(const float* __restrict__ dbl, const float* __restrict__ dtw,
                     const float* __restrict__ dtb, float* __restrict__ dt, int total) {
  const int i = blockIdx.x * blockDim.x + threadIdx.x;
  if (i >= total) return;
  const int d = i % DIC; const int bl = i / DIC;
  dt[i] = softplus_f(dbl[(size_t)bl * 33] * dtw[d] + dtb[d]);
}

// =====================================================================
// K7: SSM scan. One lane per (b,d,s); einsum over s = 16-lane shuffle
// reduction each step. dA / dBx computed on the fly (saves 150MB).
// =====================================================================
__global__ void mamba_scan(const float* __restrict__ dt,
                           const float* __restrict__ dbl,
                           const float* __restrict__ xi,
                           const float* __restrict__ A_log,
                           float* __restrict__ ysOut) {
  const int tid  = blockIdx.x * blockDim.x + threadIdx.x;  // 0..1151
  const int pair = tid >> 4;                               // b*DI + d
  const int s    = tid & 15;
  const int b    = pair / DIC;
  const int d    = pair % DIC;
  const float Ads = -__expf(A_log[d * DSC + s]);
  float h = 0.f;
  for (int l = 0; l < LLC; ++l) {
    const size_t base = (size_t)b * LLC + l;
    const float dtv = dt[base * DIC + d];
    const float xiv = xi[base * DIC + d];
    const float Bs  = dbl[base * 33 + 1 + s];
    const float Cs  = dbl[base * 33 + 17 + s];
    h = __expf(dtv * Ads) * h + dtv * Bs * xiv;
    float p = h * Cs;
    p += __shfl_xor(p, 1);
    p += __shfl_xor(p, 2);
    p += __shfl_xor(p, 4);
    p += __shfl_xor(p, 8);
    if (s == 0) ysOut[base * DIC + d] = p;
  }
}

// K8a: y = (ys + D*xi) * silu(z)
__global__ void gate_k(const float* __restrict__ ys, const float* __restrict__ xi2,
                       const float* __restrict__ xz, const float* __restrict__ Dp,
                       float* __restrict__ yact, int total) {
  const int i = blockIdx.x * blockDim.x + threadIdx.x;
  if (i >= total) return;
  const int d = i % DIC; const int bl = i / DIC;
  const float z = xz[(size_t)bl * 36 + DIC + d];
  yact[i] = (ys[i] + Dp[d] * xi2[i]) * silu_f(z);
}

// K8b: out_proj + trc_inverse + y_new, clipped row-coordinate map
__global__ void pymap_k(const float* __restrict__ yact, const float* __restrict__ opw,
                        const float* __restrict__ ycum, const float* __restrict__ altho,
                        float* __restrict__ pym, int total) {
  const int i = blockIdx.x * blockDim.x + threadIdx.x;  // (b,k,h,w)
  if (i >= total) return;
  const int v = i & 127;
  const int h = (i >> 7) & 127;
  const int kk = (i >> 14) % KKC;
  const int b  = (i >> 14) / KKC;
  const int h2 = h >> 1, t = h & 1;
  const int l = ((h2 * WWI + v) << 1) + t;
  const float* ya = yact + ((size_t)b * LLC + l) * DIC;
  const float* wp = opw + kk * DIC;
  float acc = 0.f;
  for (int d = 0; d < DIC; ++d) acc += ya[d] * wp[d];
  float wgt = fmaxf(softplus_f(altho[0]), 0.01f);
  float p = wgt * acc + (float)h + ycum[i];
  pym[i] = fminf(fmaxf(p, 0.f), (float)(HH - 1));
}

// =====================================================================
// K9: fused bilinear-sample + 9x1 strided conv (64->64) on f32 WMMA.
// Per wave: 2 M-tiles x 16 oc. Column coord always integral -> 1-D lerp
// along y only: per step 4x b64 sample loads + 1x b64 weight load for
// 2 wmma. All sampling setup hoisted to the tap loop.
// =====================================================================
__global__ void dsconv_wmma(const float* __restrict__ xt,   // [B][16384][64]
                            const float* __restrict__ pym,
                            const float* __restrict__ wt,   // [64][9][64]
                            const float* __restrict__ bias,
                            float* __restrict__ out) {
  const int lane = threadIdx.x & 31;
  const int ocb  = (threadIdx.x >> 5) * 16;
  const int p0   = blockIdx.x * 32;
  const int b    = p0 >> 14;
  const int rem0 = p0 & 16383;
  const int m    = lane & 15;
  const int kb   = (lane >> 4) * 2;
  const int oc   = ocb + (lane & 15);

  const int pix0 = rem0 + m,       pix1 = rem0 + 16 + m;
  const int u0 = pix0 >> 7, v0 = pix0 & 127;
  const int u1 = pix1 >> 7, v1 = pix1 & 127;

  const float* xb = xt + (((size_t)b) << 14) * 64;

  v8f acc0 = {}, acc1 = {};
  for (int r = 0; r < 9; ++r) {
    const float* pr = pym + ((size_t)(b * KKC + r) << 14);
    const float py0 = pr[(u0 << 7) + v0];
    const float py1 = pr[(u1 << 7) + v1];
    const float fy0 = floorf(py0), fy1 = floorf(py1);
    const float wy0 = py0 - fy0,   wy1 = py1 - fy1;
    const int y00 = clampi((int)fy0, 0, HH - 1), y01 = clampi(y00 + 1, 0, HH - 1);
    const int y10 = clampi((int)fy1, 0, HH - 1), y11 = clampi(y10 + 1, 0, HH - 1);
    const int c0  = clampi(v0 + r - 4, 0, WWI - 1);
    const int c1  = clampi(v1 + r - 4, 0, WWI - 1);
    const float* p00 = xb + (((size_t)(y00 * WWI + c0)) << 6) + kb;
    const float* p01 = xb + (((size_t)(y01 * WWI + c0)) << 6) + kb;
    const float* p10 = xb + (((size_t)(y10 * WWI + c1)) << 6) + kb;
    const float* p11 = xb + (((size_t)(y11 * WWI + c1)) << 6) + kb;
    const float* wp  = wt + ((oc * 9 + r) << 6) + kb;
    #pragma unroll 4
    for (int is = 0; is < 16; ++is) {
      v2f s00 = *(const v2f*)p00;
      v2f s01 = *(const v2f*)p01;
      v2f s10 = *(const v2f*)p10;
      v2f s11 = *(const v2f*)p11;
      v2f bv  = *(const v2f*)wp;
      v2f a0, a1;
      a0.x = s00.x * (1.f - wy0) + s01.x * wy0;
      a0.y = s00.y * (1.f - wy0) + s01.y * wy0;
      a1.x = s10.x * (1.f - wy1) + s11.x * wy1;
      a1.y = s10.y * (1.f - wy1) + s11.y * wy1;
      acc0 = wmma4(a0, bv, acc0);
      acc1 = wmma4(a1, bv, acc1);
      p00 += 4; p01 += 4; p10 += 4; p11 += 4; wp += 4;
    }
  }
  const int mb  = (lane >> 4) * 8;
  const int ocw = ocb + (lane & 15);
  const float bb = bias[ocw];
  float* ob = out + (size_t)(b * COUTC + ocw) * LLC + rem0 + mb;
  for (int j = 0; j < 8; ++j) {
    ob[j]      = acc0[j] + bb;
    ob[16 + j] = acc1[j] + bb;
  }
}

// =====================================================================
extern "C" void kernel_launch(void* const* d_in, const int* in_sizes, int n_in,
                              void* d_out, int out_size, void* d_ws, size_t ws_size,
                              hipStream_t stream) {
  const float* x        = (const float*)d_in[0];
  const float* offset_w = (const float*)d_in[1];
  const float* offset_b = (const float*)d_in[2];
  const float* gno_g    = (const float*)d_in[3];
  const float* gno_b    = (const float*)d_in[4];
  const float* altho    = (const float*)d_in[5];
  const float* in_proj_w= (const float*)d_in[6];
  const float* conv1d_w = (const float*)d_in[7];
  const float* conv1d_b = (const float*)d_in[8];
  const float* x_proj_w = (const float*)d_in[9];
  const float* dt_proj_w= (const float*)d_in[10];
  const float* dt_proj_b= (const float*)d_in[11];
  const float* A_log    = (const float*)d_in[12];
  const float* D_p      = (const float*)d_in[13];
  const float* out_proj_w=(const float*)d_in[14];
  const float* dsc_w    = (const float*)d_in[15];
  const float* dsc_b    = (const float*)d_in[16];
  const float* gn_g     = (const float*)d_in[17];
  const float* gn_b     = (const float*)d_in[18];
  float* outp = (float*)d_out;

  // ---- workspace carve-up (floats) ----
  float* ws = (float*)d_ws;
  const size_t WT1_N  = 32 * 9 * 64;                   //    18,432
  const size_t WT2_N  = 64 * 9 * 64;                   //    36,864
  const size_t XT_N   = (size_t)BSZ * LLC * CINC;      // 4,194,304
  const size_t OFF_N  = (size_t)BSZ * 2 * KKC * LLC;   // 1,179,648
  const size_t YCUM_N = (size_t)BSZ * KKC * LLC;       //   589,824
  const size_t YSEQ_N = (size_t)BSZ * LLC * DMC;       //   589,824
  const size_t XZ_N   = (size_t)BSZ * LLC * 36;        // 2,359,296
  const size_t XI2_N  = (size_t)BSZ * LLC * DIC;       // 1,179,648
  const size_t DBL_N  = (size_t)BSZ * LLC * 33;        // 2,162,688
  const size_t DT_N   = XI2_N;
  const size_t YS_N   = XI2_N;
  const size_t YACT_N = XI2_N;
  const size_t PYM_N  = YCUM_N;
  const size_t C2_N   = (size_t)BSZ * COUTC * LLC;     // 4,194,304

  float* wt1   = ws;            ws += WT1_N;
  float* wt2   = ws;            ws += WT2_N;
  float* xtb   = ws;            ws += XT_N;
  float* off   = ws;            ws += OFF_N;
  float* offt  = ws;            ws += OFF_N;
  float* ycum  = ws;            ws += YCUM_N;
  float* yseq  = ws;            ws += YSEQ_N;
  float* xz    = ws;            ws += XZ_N;
  float* xi2   = ws;            ws += XI2_N;
  float* dbl   = ws;            ws += DBL_N;
  float* dt    = ws;            ws += DT_N;
  float* ysbuf = ws;            ws += YS_N;
  float* yact  = ws;            ws += YACT_N;
  float* pym   = ws;            ws += PYM_N;
  float* c2    = ws;            ws += C2_N;

  // 0) weight re-layout + pixel-major transpose of x
  prep_w_k<<<(int)((WT2_N + 255) / 256), 256, 0, stream>>>(offset_w, dsc_w, wt1, wt2);
  transpose_x_k<<<BSZ * 256, 256, 0, stream>>>(x, xtb);
  // 1) offset conv (WMMA, 2 M-tiles/wave)
  conv3x3_wmma<<<2048, 64, 0, stream>>>(xtb, wt1, offset_b, off);
  // 2) groupnorm(K groups) + tanh
  groupnorm_k<<<BSZ * KKC, 256, 0, stream>>>(off, offt, gno_g, gno_b, KKC, 2, 1);
  // 3) cumulative offsets + sequence build
  cum_seq_k<<<(BSZ * LLC) / 256, 256, 0, stream>>>(offt, ycum, yseq);
  // 4) mamba in_proj
  {
    const int total = (int)XZ_N;
    in_proj_k<<<(total + 255) / 256, 256, 0, stream>>>(yseq, in_proj_w, xz, total);
  }
  // 5) depthwise conv1d + silu
  {
    const int total = (int)XI2_N;
    conv1d_silu_k<<<(total + 255) / 256, 256, 0, stream>>>(xz, conv1d_w, conv1d_b, xi2, total);
  }
  // 6) x_proj
  {
    const int total = (int)DBL_N;
    x_proj_k<<<(total + 255) / 256, 256, 0, stream>>>(xi2, x_proj_w, dbl, total);
  }
  // 6b) dt
  {
    const int total = (int)DT_N;
    dt_k<<<(total + 255) / 256, 256, 0, stream>>>(dbl, dt_proj_w, dt_proj_b, dt, total);
  }
  // 7) selective scan (1152 parallel recurrences)
  mamba_scan<<<9, 128, 0, stream>>>(dt, dbl, xi2, A_log, ysbuf);
  // 8a) gate
  {
    const int total = (int)YACT_N;
    gate_k<<<(total + 255) / 256, 256, 0, stream>>>(ysbuf, xi2, xz, D_p, yact, total);
  }
  // 8b) out_proj + coordinate map
  {
    const int total = (int)PYM_N;
    pymap_k<<<(total + 255) / 256, 256, 0, stream>>>(yact, out_proj_w, ycum, altho, pym, total);
  }
  // 9) fused bilinear + strided conv (WMMA, 2 M-tiles/wave)
  dsconv_wmma<<<2048, 128, 0, stream>>>(xtb, pym, wt2, dsc_b, c2);
  // 10) final groupnorm -> d_out
  groupnorm_k<<<BSZ * (COUTC / 4), 256, 0, stream>>>(c2, outp, gn_g, gn_b, COUTC / 4, 4, 0);

  (void)in_sizes; (void)n_in; (void)out_size; (void)ws_size;
}